// HiPPOWithGradients_81913616270046
// MI455X (gfx1250) — compile-verified
//
#include <hip/hip_runtime.h>
#include <hip/hip_bf16.h>
#include <stdint.h>

// HiPPO update: out = A @ s + p.flat - w * g.flat, for 3 segments.
//
// f32 GEMV on V_WMMA_F32_16X16X4_F32. Each 32-thread (1-wave) workgroup owns
// a 16-row strip of A. A 16x128 A-tile plus the matching 128-elem s chunk are
// moved memory->LDS with GLOBAL_LOAD_ASYNC_TO_LDS_B128 (ASYNCcnt-tracked,
// no VGPR staging), double-buffered so tile t+1 streams in while tile t is
// consumed by 32 chained WMMAs. Column 0 of the WMMA-B operand carries
// s[k..k+3]; all other columns are zeroed so D[:,1..15] stay 0 and D[:,0]
// accumulates the 16 dot products for the strip.

typedef float v2f __attribute__((ext_vector_type(2)));
typedef float v8f __attribute__((ext_vector_type(8)));

#define TK 128
#define LDS_STRIDE (TK + 4)   // pad: column reads hit distinct banks

struct Seg {
    const float* A;    // n x n row-major
    const float* s;    // n
    const float* p;    // n (flattened param)
    const float* g;    // n (flattened grad)
    float*       out;  // n
    int          n;
    int          nstrips; // n / 16
};

__global__ __launch_bounds__(32)
void hippo_wmma_gemv(Seg s0, Seg s1, Seg s2, const float* __restrict__ wptr)
{
    __shared__ float lds_a[2][16 * LDS_STRIDE];
    __shared__ float lds_s[2][TK];

    int bid = (int)blockIdx.x;
    Seg S;
    if (bid < s0.nstrips) {
        S = s0;
    } else if (bid < s0.nstrips + s1.nstrips) {
        S = s1; bid -= s0.nstrips;
    } else {
        S = s2; bid -= s0.nstrips + s1.nstrips;
    }

    const int n     = S.n;
    const int r0    = bid * 16;              // first output row of this strip
    const int lane  = (int)threadIdx.x;      // 0..31 (wave32)
    const int mrow  = lane & 15;             // M row this lane serves in WMMA-A
    const int khalf = lane >> 4;             // 0 -> K{0,1}, 1 -> K{2,3}

    const float w = wptr[0];

    // LDS byte offsets (generic-pointer low 32 bits == LDS offset).
    const uint32_t a_addr0 = (uint32_t)(uintptr_t)&lds_a[0][0];
    const uint32_t a_addr1 = (uint32_t)(uintptr_t)&lds_a[1][0];
    const uint32_t s_addr0 = (uint32_t)(uintptr_t)&lds_s[0][0];
    const uint32_t s_addr1 = (uint32_t)(uintptr_t)&lds_s[1][0];

    // Issue 17 async b128 copies for tile t into buffer bufi:
    //   1 for the s chunk, 16 covering the 16x128 A tile (coalesced rows).
    auto issue_tile = [&](int t, int bufi) {
        const int k0 = t * TK;
        const uint32_t a_base = bufi ? a_addr1 : a_addr0;
        const uint32_t s_base = bufi ? s_addr1 : s_addr0;

        const uint64_t g_s = (uint64_t)(uintptr_t)(S.s + k0 + lane * 4);
        const uint32_t l_s = s_base + (uint32_t)(lane * 16);
        asm volatile("global_load_async_to_lds_b128 %0, %1, off"
                     :: "v"(l_s), "v"(g_s) : "memory");

        #pragma unroll
        for (int it = 0; it < 16; ++it) {
            const int idx = it * 32 + lane;      // 0..511 float4 slots
            const int row = idx >> 5;            // / (TK/4)
            const int c4  = idx & 31;
            const uint64_t g_a = (uint64_t)(uintptr_t)
                (S.A + (size_t)(r0 + row) * (size_t)n + k0 + c4 * 4);
            const uint32_t l_a = a_base + (uint32_t)((row * LDS_STRIDE + c4 * 4) * 4);
            asm volatile("global_load_async_to_lds_b128 %0, %1, off"
                         :: "v"(l_a), "v"(g_a) : "memory");
        }
    };

    v8f acc = {0.f, 0.f, 0.f, 0.f, 0.f, 0.f, 0.f, 0.f};

    const int ntiles = n / TK;

    issue_tile(0, 0);

    for (int t = 0; t < ntiles; ++t) {
        if (t + 1 < ntiles) {
            issue_tile(t + 1, (t + 1) & 1);
            // tile t's 17 asyncs precede tile t+1's 17; in-order completion
            // means ASYNCcnt <= 17 implies tile t fully landed in LDS.
            asm volatile("s_wait_asynccnt 0x11" ::: "memory");
        } else {
            asm volatile("s_wait_asynccnt 0x0" ::: "memory");
        }

        const float* la = &lds_a[t & 1][0];
        const float* ls = &lds_s[t & 1][0];

        #pragma unroll 8
        for (int kk = 0; kk < TK; kk += 4) {
            // WMMA-A f32 16x4: lanes 0-15 hold A[M=lane][kk+{0,1}],
            //                  lanes 16-31 hold A[M=lane-16][kk+{2,3}]
            v2f av = *(const v2f*)(la + mrow * LDS_STRIDE + kk + khalf * 2);

            // WMMA-B f32 4x16: column 0 (lanes 0 and 16) carries the s chunk.
            v2f sv = *(const v2f*)(ls + kk + khalf * 2);
            const bool col0 = (mrow == 0);
            v2f bv;
            bv.x = col0 ? sv.x : 0.0f;
            bv.y = col0 ? sv.y : 0.0f;

            acc = __builtin_amdgcn_wmma_f32_16x16x4_f32(
                /*neg_a=*/false, av, /*neg_b=*/false, bv,
                /*c_mod=*/(short)0, acc, /*reuse_a=*/false, /*reuse_b=*/false);
        }

        // Ensure this tile's LDS reads retired before the buffer is
        // overwritten by the async batch (tile t+2) issued in the next iteration.
        asm volatile("s_wait_dscnt 0x0" ::: "memory");
    }

    // ---- fused epilogue: D[:,0] lives in lanes 0 (M=0..7) and 16 (M=8..15) ----
    if (mrow == 0) {
        const int rbase = r0 + khalf * 8;
        #pragma unroll
        for (int j = 0; j < 8; ++j) {
            const int r = rbase + j;
            S.out[r] = acc[j] + S.p[r] - w * S.g[r];
        }
    }
}

extern "C" void kernel_launch(void* const* d_in, const int* in_sizes, int n_in,
                              void* d_out, int out_size, void* d_ws, size_t ws_size,
                              hipStream_t stream) {
    (void)in_sizes; (void)n_in; (void)d_ws; (void)ws_size; (void)out_size;

    // setup_inputs() dict order:
    // 0:param0 1:grad0 2:state0 3:A0  4:param1 5:grad1 6:state1 7:A1
    // 8:param2 9:grad2 10:state2 11:A2  12:weights
    float* out = (float*)d_out;

    Seg s0, s1, s2;
    s0.A = (const float*)d_in[3];  s0.s = (const float*)d_in[2];
    s0.p = (const float*)d_in[0];  s0.g = (const float*)d_in[1];
    s0.out = out;                  s0.n = 8192; s0.nstrips = 8192 / 16;

    s1.A = (const float*)d_in[7];  s1.s = (const float*)d_in[6];
    s1.p = (const float*)d_in[4];  s1.g = (const float*)d_in[5];
    s1.out = out + 8192;           s1.n = 8192; s1.nstrips = 8192 / 16;

    s2.A = (const float*)d_in[11]; s2.s = (const float*)d_in[10];
    s2.p = (const float*)d_in[8];  s2.g = (const float*)d_in[9];
    s2.out = out + 16384;          s2.n = 4096; s2.nstrips = 4096 / 16;

    const float* wptr = (const float*)d_in[12];

    const int nblocks = s0.nstrips + s1.nstrips + s2.nstrips; // 1280
    hippo_wmma_gemv<<<nblocks, 32, 0, stream>>>(s0, s1, s2, wptr);
}